// DecoderSplattingCUDA_1340029796677
// MI455X (gfx1250) — compile-verified
//
#include <hip/hip_runtime.h>
#include <math.h>

typedef float v2f __attribute__((ext_vector_type(2)));
typedef float v8f __attribute__((ext_vector_type(8)));

// GCC-vector int types matching the async-LDS builtin signatures
// (diagnostic showed: "__attribute__((__vector_size__(4*sizeof(int)))) int __device__ *")
typedef int vi4 __attribute__((vector_size(16)));
typedef int vi2 __attribute__((vector_size(8)));
typedef __attribute__((address_space(1))) vi4* g4p_t;   // global (AS1)
typedef __attribute__((address_space(3))) vi4* s4p_t;   // LDS    (AS3)
typedef __attribute__((address_space(1))) vi2* g2p_t;
typedef __attribute__((address_space(3))) vi2* s2p_t;

#define SH_C0 0.28209479177387814f
#define ZNEAR 0.1f
#define ZFAR  1000.0f

#if defined(__has_builtin)
#  if __has_builtin(__builtin_amdgcn_global_load_async_to_lds_b128) && \
      __has_builtin(__builtin_amdgcn_global_load_async_to_lds_b64)
#    define USE_ASYNC_BUILTIN 1
#  else
#    define USE_ASYNC_BUILTIN 0
#  endif
#  if __has_builtin(__builtin_amdgcn_s_wait_asynccnt)
#    define WAIT_ASYNC(n) __builtin_amdgcn_s_wait_asynccnt(n)
#  else
#    define WAIT_ASYNC(n) asm volatile("s_wait_asynccnt %0" :: "i"(n) : "memory")
#  endif
#else
#  define USE_ASYNC_BUILTIN 0
#  define WAIT_ASYNC(n) asm volatile("s_wait_asynccnt %0" :: "i"(n) : "memory")
#endif

// ---------------------------------------------------------------------------
// Generic 4x4 inverse (adjugate), one thread per camera (C is tiny).
// ---------------------------------------------------------------------------
__device__ inline void invert4x4(const float* m, float* invOut) {
    float inv[16];
    inv[0]  =  m[5]*m[10]*m[15] - m[5]*m[11]*m[14] - m[9]*m[6]*m[15] + m[9]*m[7]*m[14] + m[13]*m[6]*m[11] - m[13]*m[7]*m[10];
    inv[4]  = -m[4]*m[10]*m[15] + m[4]*m[11]*m[14] + m[8]*m[6]*m[15] - m[8]*m[7]*m[14] - m[12]*m[6]*m[11] + m[12]*m[7]*m[10];
    inv[8]  =  m[4]*m[9]*m[15]  - m[4]*m[11]*m[13] - m[8]*m[5]*m[15] + m[8]*m[7]*m[13] + m[12]*m[5]*m[11] - m[12]*m[7]*m[9];
    inv[12] = -m[4]*m[9]*m[14]  + m[4]*m[10]*m[13] + m[8]*m[5]*m[14] - m[8]*m[6]*m[13] - m[12]*m[5]*m[10] + m[12]*m[6]*m[9];
    inv[1]  = -m[1]*m[10]*m[15] + m[1]*m[11]*m[14] + m[9]*m[2]*m[15] - m[9]*m[3]*m[14] - m[13]*m[2]*m[11] + m[13]*m[3]*m[10];
    inv[5]  =  m[0]*m[10]*m[15] - m[0]*m[11]*m[14] - m[8]*m[2]*m[15] + m[8]*m[3]*m[14] + m[12]*m[2]*m[11] - m[12]*m[3]*m[10];
    inv[9]  = -m[0]*m[9]*m[15]  + m[0]*m[11]*m[13] + m[8]*m[1]*m[15] - m[8]*m[3]*m[13] - m[12]*m[1]*m[11] + m[12]*m[3]*m[9];
    inv[13] =  m[0]*m[9]*m[14]  - m[0]*m[10]*m[13] - m[8]*m[1]*m[14] + m[8]*m[2]*m[13] + m[12]*m[1]*m[10] - m[12]*m[2]*m[9];
    inv[2]  =  m[1]*m[6]*m[15]  - m[1]*m[7]*m[14]  - m[5]*m[2]*m[15] + m[5]*m[3]*m[14] + m[13]*m[2]*m[7]  - m[13]*m[3]*m[6];
    inv[6]  = -m[0]*m[6]*m[15]  + m[0]*m[7]*m[14]  + m[4]*m[2]*m[15] - m[4]*m[3]*m[14] - m[12]*m[2]*m[7]  + m[12]*m[3]*m[6];
    inv[10] =  m[0]*m[5]*m[15]  - m[0]*m[7]*m[13]  - m[4]*m[1]*m[15] + m[4]*m[3]*m[13] + m[12]*m[1]*m[7]  - m[12]*m[3]*m[5];
    inv[14] = -m[0]*m[5]*m[14]  + m[0]*m[6]*m[13]  + m[4]*m[1]*m[14] - m[4]*m[2]*m[13] - m[12]*m[1]*m[6]  + m[12]*m[2]*m[5];
    inv[3]  = -m[1]*m[6]*m[11]  + m[1]*m[7]*m[10]  + m[5]*m[2]*m[11] - m[5]*m[3]*m[10] - m[9]*m[2]*m[7]   + m[9]*m[3]*m[6];
    inv[7]  =  m[0]*m[6]*m[11]  - m[0]*m[7]*m[10]  - m[4]*m[2]*m[11] + m[4]*m[3]*m[10] + m[8]*m[2]*m[7]   - m[8]*m[3]*m[6];
    inv[11] = -m[0]*m[5]*m[11]  + m[0]*m[7]*m[9]   + m[4]*m[1]*m[11] - m[4]*m[3]*m[9]  - m[8]*m[1]*m[7]   + m[8]*m[3]*m[5];
    inv[15] =  m[0]*m[5]*m[10]  - m[0]*m[6]*m[9]   - m[4]*m[1]*m[10] + m[4]*m[2]*m[9]  + m[8]*m[1]*m[6]   - m[8]*m[2]*m[5];
    float det = m[0]*inv[0] + m[1]*inv[4] + m[2]*inv[8] + m[3]*inv[12];
    float idet = 1.0f / det;
    for (int i = 0; i < 16; i++) invOut[i] = inv[i] * idet;
}

// ---------------------------------------------------------------------------
// Kernel 1: per-camera view matrix + intrinsics.
// camParams[c*32]: R(9) row-major, t(3), fx, fy, cx, cy
// ---------------------------------------------------------------------------
__global__ void k_camera(const float* __restrict__ base,
                         const float* __restrict__ target,
                         const float* __restrict__ K,
                         int b, int v, float* __restrict__ camParams) {
    int c = blockIdx.x * blockDim.x + threadIdx.x;
    int C = b * v;
    if (c >= C) return;
    int bi = c / v;
    float invB[16], extr[16], view[16];
    invert4x4(base + bi * 16, invB);
    const float* tp = target + c * 16;
    for (int i = 0; i < 4; i++)
        for (int j = 0; j < 4; j++) {
            float s = 0.f;
            for (int k = 0; k < 4; k++) s += invB[i*4+k] * tp[k*4+j];
            extr[i*4+j] = s;
        }
    invert4x4(extr, view);
    float* cp = camParams + c * 32;
    for (int i = 0; i < 3; i++) {
        for (int j = 0; j < 3; j++) cp[i*3+j] = view[i*4+j];
        cp[9 + i] = view[i*4+3];
    }
    const float* kk = K + c * 9;
    cp[12] = kk[0];  // fx   (Kn row-normalize then *W/H cancels)
    cp[13] = kk[4];  // fy
    cp[14] = kk[2];  // cx
    cp[15] = kk[5];  // cy
}

// ---------------------------------------------------------------------------
// Kernel 2: project each gaussian; emit z, rank-6 bilinear coefficients
// (padded to 8), and meta {effective opacity, rgb}.
//   power(p,g) = [1, px, py, px^2, py^2, px*py] . [c0..c5]
// ---------------------------------------------------------------------------
__global__ void k_project(const float* __restrict__ m1, const float* __restrict__ c1,
                          const float* __restrict__ s1, const float* __restrict__ o1,
                          const float* __restrict__ m2, const float* __restrict__ c2,
                          const float* __restrict__ s2, const float* __restrict__ o2,
                          const float* __restrict__ camParams,
                          int b, int v, int Npart, int G, int Gp,
                          float* __restrict__ zbuf, float* __restrict__ coef,
                          float* __restrict__ meta) {
    int idx = blockIdx.x * blockDim.x + threadIdx.x;
    int C = b * v;
    if (idx >= C * G) return;
    int c = idx / G, g = idx % G;
    int bi = c / v;
    const float *mp, *cvp, *sp, *op;
    if (g < Npart) {
        int li = bi * Npart + g;
        mp = m1 + li*3; cvp = c1 + li*9; sp = s1 + li*3; op = o1 + li;
    } else {
        int li = bi * Npart + (g - Npart);
        mp = m2 + li*3; cvp = c2 + li*9; sp = s2 + li*3; op = o2 + li;
    }
    const float* cam = camParams + c * 32;
    float R[9];
    #pragma unroll
    for (int i = 0; i < 9; i++) R[i] = cam[i];
    float tx = cam[9], ty = cam[10], tz = cam[11];
    float fx = cam[12], fy = cam[13], cx = cam[14], cy = cam[15];

    float X = mp[0], Y = mp[1], Z = mp[2];
    float px_ = R[0]*X + R[1]*Y + R[2]*Z + tx;
    float py_ = R[3]*X + R[4]*Y + R[5]*Z + ty;
    float pz_ = R[6]*X + R[7]*Y + R[8]*Z + tz;
    float zc = fmaxf(pz_, 1e-6f);
    float u  = fx * px_ / zc + cx;
    float vv = fy * py_ / zc + cy;

    // cov_cam = R S R^T
    float S[9];
    #pragma unroll
    for (int i = 0; i < 9; i++) S[i] = cvp[i];
    float RS[9];
    #pragma unroll
    for (int i = 0; i < 3; i++)
        #pragma unroll
        for (int j = 0; j < 3; j++) {
            float s = 0.f;
            #pragma unroll
            for (int k = 0; k < 3; k++) s += R[i*3+k] * S[k*3+j];
            RS[i*3+j] = s;
        }
    float Cc[9];
    #pragma unroll
    for (int i = 0; i < 3; i++)
        #pragma unroll
        for (int j = 0; j < 3; j++) {
            float s = 0.f;
            #pragma unroll
            for (int k = 0; k < 3; k++) s += RS[i*3+k] * R[j*3+k];
            Cc[i*3+j] = s;
        }

    float j00 = fx / zc,        j02 = -fx * px_ / (zc * zc);
    float j11 = fy / zc,        j12 = -fy * py_ / (zc * zc);
    float t00 = j00*Cc[0] + j02*Cc[6];
    float t01 = j00*Cc[1] + j02*Cc[7];
    float t02 = j00*Cc[2] + j02*Cc[8];
    float t11 = j11*Cc[4] + j12*Cc[7];
    float t12 = j11*Cc[5] + j12*Cc[8];
    float a   = t00*j00 + t02*j02 + 0.3f;
    float bq  = t01*j11 + t02*j12;
    float d   = t11*j11 + t12*j12 + 0.3f;
    float det = a*d - bq*bq;
    bool valid = (pz_ > ZNEAR) && (pz_ < ZFAR) && (det > 0.f);
    float inv_det = 1.f / det;
    float ca =  d * inv_det;
    float cb = -bq * inv_det;
    float cc =  a * inv_det;
    if (!valid) { ca = cb = cc = 0.f; u = 0.f; vv = 0.f; }   // keep WMMA tile NaN-free

    float r_ = fmaxf(SH_C0 * sp[0] + 0.5f, 0.f);
    float g_ = fmaxf(SH_C0 * sp[1] + 0.5f, 0.f);
    float b_ = fmaxf(SH_C0 * sp[2] + 0.5f, 0.f);
    float opv = valid ? op[0] : 0.f;

    float* cf = coef + (size_t)(c * Gp + g) * 8;
    cf[0] = -0.5f*ca*u*u - 0.5f*cc*vv*vv - cb*u*vv;  // * 1
    cf[1] = ca*u + cb*vv;                            // * px
    cf[2] = cc*vv + cb*u;                            // * py
    cf[3] = -0.5f*ca;                                // * px^2
    cf[4] = -0.5f*cc;                                // * py^2
    cf[5] = -cb;                                     // * px*py
    cf[6] = 0.f; cf[7] = 0.f;
    float* mt = meta + (size_t)(c * Gp + g) * 4;
    mt[0] = opv; mt[1] = r_; mt[2] = g_; mt[3] = b_;
    zbuf[c * Gp + g] = pz_;
}

// ---------------------------------------------------------------------------
// Kernel 3: deterministic stable rank-sort by z + scatter into sorted SoA.
// Pads [G, Gp) with zeros (alpha = 0).
// ---------------------------------------------------------------------------
__global__ void k_sort(const float* __restrict__ zbuf,
                       const float* __restrict__ coef,
                       const float* __restrict__ meta,
                       int C, int G, int Gp,
                       float* __restrict__ scoef, float* __restrict__ smeta) {
    int idx = blockIdx.x * blockDim.x + threadIdx.x;
    if (idx >= C * Gp) return;
    int c = idx / Gp, g = idx % Gp;
    if (g >= G) {
        float* sc = scoef + (size_t)(c * Gp + g) * 8;
        #pragma unroll
        for (int i = 0; i < 8; i++) sc[i] = 0.f;
        float* sm = smeta + (size_t)(c * Gp + g) * 4;
        #pragma unroll
        for (int i = 0; i < 4; i++) sm[i] = 0.f;
        return;
    }
    const float* z = zbuf + c * Gp;
    float zg = z[g];
    int rank = 0;
    for (int j = 0; j < G; j++) {
        float zj = z[j];
        rank += (zj < zg) || (zj == zg && j < g);
    }
    const float* cf = coef + (size_t)(c * Gp + g) * 8;
    float* sc = scoef + (size_t)(c * Gp + rank) * 8;
    #pragma unroll
    for (int i = 0; i < 8; i++) sc[i] = cf[i];
    const float* mt = meta + (size_t)(c * Gp + g) * 4;
    float* sm = smeta + (size_t)(c * Gp + rank) * 4;
    #pragma unroll
    for (int i = 0; i < 4; i++) sm[i] = mt[i];
}

// ---------------------------------------------------------------------------
// Async staging of one 16-gaussian chunk (512B coef + 256B meta) into LDS.
// Wave-wide: 32 lanes x b128 (coef) + 32 lanes x b64 (meta). Tracked on
// ASYNCcnt (2 ops per chunk).
// ---------------------------------------------------------------------------
__device__ __forceinline__ void issue_chunk(const float* cbase, const float* mbase,
                                            float* wbuf, int ch, int lane) {
    const float* gc = cbase + (size_t)ch * 128 + lane * 4;   // 16B per lane
    const float* gm = mbase + (size_t)ch * 64  + lane * 2;   // 8B per lane
    float* lc = wbuf + (ch & 1) * 192 + lane * 4;
    float* lm = wbuf + (ch & 1) * 192 + 128 + lane * 2;
#if USE_ASYNC_BUILTIN
    __builtin_amdgcn_global_load_async_to_lds_b128(
        (g4p_t)(uintptr_t)gc, (s4p_t)(uint32_t)(uintptr_t)lc, 0, 0);
    __builtin_amdgcn_global_load_async_to_lds_b64(
        (g2p_t)(uintptr_t)gm, (s2p_t)(uint32_t)(uintptr_t)lm, 0, 0);
#else
    unsigned lco = (unsigned)(uintptr_t)lc;
    unsigned lmo = (unsigned)(uintptr_t)lm;
    asm volatile("global_load_async_to_lds_b128 %0, %1, off\n\t"
                 "global_load_async_to_lds_b64 %2, %3, off"
                 :: "v"(lco), "v"(gc), "v"(lmo), "v"(gm)
                 : "memory");
#endif
}

// ---------------------------------------------------------------------------
// Kernel 4: render. One wave32 per 16-pixel tile. Per 16-gaussian chunk:
//  - chunk data double-buffered in LDS via GLOBAL_LOAD_ASYNC_TO_LDS_* /
//    s_wait_asynccnt, next chunk's DMA overlapped with current compute;
//  - two V_WMMA_F32_16X16X4_F32 compute the 16x16 power tile (K 6->8 pad);
//  - each lane composites its 8 sorted gaussians, __shfl_xor(16) stitches
//    the two lane-halves in depth order.
// D layout: lanes L / L+16 hold pixel N=L; VGPR m = gaussian M=m (+8 hi half).
// A 16x4 layout: lanes 0-15 -> {K0,K1}, lanes 16-31 -> {K2,K3}; B mirrors it.
// ---------------------------------------------------------------------------
__global__ void __launch_bounds__(256) k_render(
        const float* __restrict__ scoef, const float* __restrict__ smeta,
        const float* __restrict__ bg,
        int C, int Gp, int W, int HW, int tilesPerCam,
        float* __restrict__ out) {
    __shared__ float lds[8 * 2 * 192];   // 8 waves x double buffer x 192 floats
    int tid = blockIdx.x * blockDim.x + threadIdx.x;
    int wave = tid >> 5;
    int lane = tid & 31;
    if (wave >= C * tilesPerCam) return;   // wave-uniform: EXEC stays all-ones
    int c    = wave / tilesPerCam;
    int tile = wave % tilesPerCam;
    int ln   = lane & 15;
    bool hi  = lane >= 16;
    int p    = tile * 16 + ln;
    float pxc = (float)(p % W) + 0.5f;
    float pyc = (float)(p / W) + 0.5f;

    // Pixel basis B (K x N): lanes 0-15 hold K0,K1 ; lanes 16-31 hold K2,K3.
    v2f B1, B2;
    if (!hi) { B1.x = 1.f; B1.y = pxc;     B2.x = pyc*pyc; B2.y = pxc*pyc; }
    else     { B1.x = pyc; B1.y = pxc*pxc; B2.x = 0.f;     B2.y = 0.f;     }

    const float* cbase = scoef + (size_t)c * Gp * 8;
    const float* mbase = smeta + (size_t)c * Gp * 4;
    float* wbuf = lds + (threadIdx.x >> 5) * 384;

    int nChunks = Gp >> 4;
    issue_chunk(cbase, mbase, wbuf, 0, lane);     // prologue: stage chunk 0

    float T = 1.f, cr = 0.f, cg = 0.f, cb = 0.f;
    for (int ch = 0; ch < nChunks; ++ch) {
        if (ch + 1 < nChunks) {
            issue_chunk(cbase, mbase, wbuf, ch + 1, lane);
            WAIT_ASYNC(2);                        // current chunk's DMA done
        } else {
            WAIT_ASYNC(0);
        }
        asm volatile("" ::: "memory");            // order LDS reads after wait

        const float* bufC = wbuf + (ch & 1) * 192;
        const float* bufM = bufC + 128;

        // A matrix (gaussian coefficients) from LDS, M = gaussian in chunk.
        const float* cf = bufC + ln * 8;
        v2f A1, A2;
        if (!hi) { A1 = *(const v2f*)(cf + 0); A2 = *(const v2f*)(cf + 4); }
        else     { A1 = *(const v2f*)(cf + 2); A2.x = 0.f; A2.y = 0.f;     }

        v8f acc = {0.f, 0.f, 0.f, 0.f, 0.f, 0.f, 0.f, 0.f};
        acc = __builtin_amdgcn_wmma_f32_16x16x4_f32(false, A1, false, B1,
                                                    (short)0, acc, false, false);
        acc = __builtin_amdgcn_wmma_f32_16x16x4_f32(false, A2, false, B2,
                                                    (short)0, acc, false, false);

        // Local front-to-back composite over this lane's 8 gaussians.
        float lt = 1.f, lr = 0.f, lg = 0.f, lb = 0.f;
        const float* mbat = bufM + (hi ? 32 : 0);
        #pragma unroll
        for (int i = 0; i < 8; i++) {
            float4 mt = *(const float4*)(mbat + i * 4);
            float power = acc[i];
            float alpha = fminf(0.99f, mt.x * __expf(power));
            alpha = ((power <= 0.f) && (alpha >= 1.f / 255.f)) ? alpha : 0.f;
            float w = alpha * lt;
            lr += w * mt.y;
            lg += w * mt.z;
            lb += w * mt.w;
            lt *= (1.f - alpha);
        }

        // Stitch lane halves: low half is nearer in depth.
        float ot  = __shfl_xor(lt, 16, 32);
        float orr = __shfl_xor(lr, 16, 32);
        float og  = __shfl_xor(lg, 16, 32);
        float ob  = __shfl_xor(lb, 16, 32);
        float lowT  = hi ? ot  : lt;
        float lowR  = hi ? orr : lr;
        float lowG  = hi ? og  : lg;
        float lowB  = hi ? ob  : lb;
        float highT = hi ? lt  : ot;
        float highR = hi ? lr  : orr;
        float highG = hi ? lg  : og;
        float highB = hi ? lb  : ob;

        cr += T * (lowR + lowT * highR);
        cg += T * (lowG + lowT * highG);
        cb += T * (lowB + lowT * highB);
        T  *= lowT * highT;
    }

    if (!hi) {
        float b0 = bg[0], b1 = bg[1], b2 = bg[2];
        out[(size_t)(c * 3 + 0) * HW + p] = cr + T * b0;
        out[(size_t)(c * 3 + 1) * HW + p] = cg + T * b1;
        out[(size_t)(c * 3 + 2) * HW + p] = cb + T * b2;
    }
}

// ---------------------------------------------------------------------------
// Host launcher
// ---------------------------------------------------------------------------
extern "C" void kernel_launch(void* const* d_in, const int* in_sizes, int n_in,
                              void* d_out, int out_size, void* d_ws, size_t ws_size,
                              hipStream_t stream) {
    const float* base   = (const float*)d_in[0];
    const float* target = (const float*)d_in[1];
    const float* K      = (const float*)d_in[2];
    const float* m1 = (const float*)d_in[3];
    const float* c1 = (const float*)d_in[4];
    const float* s1 = (const float*)d_in[5];
    const float* o1 = (const float*)d_in[6];
    const float* m2 = (const float*)d_in[7];
    const float* c2 = (const float*)d_in[8];
    const float* s2 = (const float*)d_in[9];
    const float* o2 = (const float*)d_in[10];
    const float* bg = (const float*)d_in[11];

    int b = in_sizes[0] / 16;                 // base_pose (b,4,4)
    int v = in_sizes[1] / (16 * b);           // target_pose (b,v,4,4)
    int C = b * v;
    int Npart = in_sizes[3] / (3 * b);        // means1 (b,hc,wc,3)
    int G  = 2 * Npart;
    int Gp = (G + 15) & ~15;
    int HW = out_size / (C * 3);              // output (b,v,3,H,W)
    int W  = (int)(sqrtf((float)HW) + 0.5f);  // square frames (H == W)
    int tilesPerCam = HW / 16;

    float* ws        = (float*)d_ws;
    float* camParams = ws;                                   // C*32
    float* zbuf      = camParams + (size_t)C * 32;           // C*Gp
    float* coef      = zbuf      + (size_t)C * Gp;           // C*Gp*8
    float* meta      = coef      + (size_t)C * Gp * 8;       // C*Gp*4
    float* scoef     = meta      + (size_t)C * Gp * 4;       // C*Gp*8
    float* smeta     = scoef     + (size_t)C * Gp * 8;       // C*Gp*4

    k_camera<<<(C + 63) / 64, 64, 0, stream>>>(base, target, K, b, v, camParams);

    int n2 = C * G;
    k_project<<<(n2 + 255) / 256, 256, 0, stream>>>(
        m1, c1, s1, o1, m2, c2, s2, o2, camParams,
        b, v, Npart, G, Gp, zbuf, coef, meta);

    int n3 = C * Gp;
    k_sort<<<(n3 + 255) / 256, 256, 0, stream>>>(
        zbuf, coef, meta, C, G, Gp, scoef, smeta);

    int n4 = C * tilesPerCam * 32;
    k_render<<<(n4 + 255) / 256, 256, 0, stream>>>(
        scoef, smeta, bg, C, Gp, W, HW, tilesPerCam, (float*)d_out);
}